// WCVADecoder_21698174780142
// MI455X (gfx1250) — compile-verified
//
#include <hip/hip_runtime.h>

#define DET_LEN   1024
#define T_TOT     3072
#define NSTATES   64
#define BATCH     256
#define CLIP_INIT 50.0f
#define CLAMP_V   50.0f
#define SOFT_T    1025   // out_probs[1023:2048] -> 1025 steps

// ---------------------------------------------------------------------------
// CDNA5 async global->LDS staging (ASYNCcnt path, CDNA5 ISA)
// GV mode: per-lane 64-bit global address, per-lane 32-bit LDS byte offset.
// ---------------------------------------------------------------------------
__device__ __forceinline__ void async_load_lds_b128(unsigned int lds_off,
                                                    const void* gaddr) {
    unsigned long long ga = (unsigned long long)(uintptr_t)gaddr;
    asm volatile("global_load_async_to_lds_b128 %0, %1, off"
                 :: "v"(lds_off), "v"(ga)
                 : "memory");
}
__device__ __forceinline__ void wait_asynccnt0() {
    asm volatile("s_wait_asynccnt 0" ::: "memory");
}

// Trellis edge signs: state n, predecessor idx -> (sign_r0, sign_r1), each ±1.
// Derived from CODE_GM = [[1,0,1,1,0,1,1],[1,1,1,1,0,0,1]].
__device__ __forceinline__ float2 edge_signs(int n, int idx) {
    int p = (n >> 1) + idx * 32;
    int b = n & 1;
    int c0 = b ^ ((p >> 1) & 1) ^ ((p >> 2) & 1) ^ ((p >> 4) & 1) ^ ((p >> 5) & 1);
    int c1 = b ^ (p & 1) ^ ((p >> 1) & 1) ^ ((p >> 2) & 1) ^ ((p >> 5) & 1);
    return make_float2(1.0f - 2.0f * (float)c0, 1.0f - 2.0f * (float)c1);
}

__global__ __launch_bounds__(32) void wcva_decode_kernel(
    const float* __restrict__ x,       // (BATCH, 2048)
    const float* __restrict__ wts,     // (T_TOT,)
    float* __restrict__ dec_out,       // (BATCH, 1024)
    float* __restrict__ soft_out)      // (BATCH, 1025*64)
{
    __shared__ float        xsh[2 * DET_LEN];        // 8 KB  LLR row
    __shared__ float        wsh[T_TOT];              // 12 KB weights
    __shared__ unsigned int msk[2048 * 2];           // 16 KB decision ballots t in [1024,3071]

    const int lane = threadIdx.x;   // 0..31, one wave32 per batch element
    const int b    = blockIdx.x;

    // ---- Stage inputs into LDS with the async-tensor engine ----------------
    {
        const char* xrow = (const char*)(x + (size_t)b * 2 * DET_LEN);
        unsigned int xbase = (unsigned int)(uintptr_t)(&xsh[0]);
        unsigned int wbase = (unsigned int)(uintptr_t)(&wsh[0]);
#pragma unroll
        for (int i = 0; i < 16; ++i) {              // 16 * 512B = 8 KB
            unsigned int off = (unsigned int)(i * 512 + lane * 16);
            async_load_lds_b128(xbase + off, xrow + off);
        }
#pragma unroll
        for (int i = 0; i < 24; ++i) {              // 24 * 512B = 12 KB
            unsigned int off = (unsigned int)(i * 512 + lane * 16);
            async_load_lds_b128(wbase + off, (const char*)wts + off);
        }
        wait_asynccnt0();
        __syncthreads();
    }

    // ---- Per-lane constant trellis data ------------------------------------
    // Lane L owns states sL = L and sH = L + 32.
    // preds(s) = {s>>1, (s>>1)+32}: both metrics live in lane (s>>1) [lo half]
    // or lane (s>>1)+16 [hi half] under this mapping.
    const int sL = lane, sH = lane + 32;
    const float2 gL0 = edge_signs(sL, 0), gL1 = edge_signs(sL, 1);
    const float2 gH0 = edge_signs(sH, 0), gH1 = edge_signs(sH, 1);
    // bm = l0*s0 + l1*s1 with s in {±1}:  bm = ±(l0+l1) or ±(l0-l1)
    const bool dL0 = (gL0.x != gL0.y), nL0 = (gL0.x < 0.0f);
    const bool dL1 = (gL1.x != gL1.y), nL1 = (gL1.x < 0.0f);
    const bool dH0 = (gH0.x != gH0.y), nH0 = (gH0.x < 0.0f);
    const bool dH1 = (gH1.x != gH1.y), nH1 = (gH1.x < 0.0f);
    const int srcA = lane >> 1;        // source lane for state sL's two preds
    const int srcB = srcA + 16;        // source lane for state sH's two preds

    float plo = (lane == 0) ? CLIP_INIT : 0.0f;   // in_prob0[:,0] = 50
    float phi = 0.0f;

    float* softb = soft_out + (size_t)b * (SOFT_T * NSTATES);

    // ---- Forward ACS scan over T=3072 steps --------------------------------
    for (int t = 0; t < T_TOT; ++t) {
        const int tm = t & (DET_LEN - 1);          // replication tiling
        const float l0 = xsh[2 * tm];
        const float l1 = xsh[2 * tm + 1];
        const float wt = wsh[t];

        const float S = l0 + l1;                   // exact sign-combos of llr
        const float D = l0 - l1;
        float bmL0 = dL0 ? D : S;  bmL0 = nL0 ? -bmL0 : bmL0;
        float bmL1 = dL1 ? D : S;  bmL1 = nL1 ? -bmL1 : bmL1;
        float bmH0 = dH0 ? D : S;  bmH0 = nH0 ? -bmH0 : bmH0;
        float bmH1 = dH1 ? D : S;  bmH1 = nH1 ? -bmH1 : bmH1;

        // cross-lane gather of predecessor metrics (ds_bpermute, wave32)
        const float aLo = __shfl(plo, srcA, 32);   // prob[sL>>1]
        const float aHi = __shfl(phi, srcA, 32);   // prob[(sL>>1)+32]
        const float bLo = __shfl(plo, srcB, 32);   // prob[sH>>1]
        const float bHi = __shfl(phi, srcB, 32);   // prob[(sH>>1)+32]

        // match reference rounding: cand = prob + w*bm (no fma contraction)
        const float c0 = __fadd_rn(aLo, __fmul_rn(wt, bmL0));
        const float c1 = __fadd_rn(aHi, __fmul_rn(wt, bmL1));
        const float c2 = __fadd_rn(bLo, __fmul_rn(wt, bmH0));
        const float c3 = __fadd_rn(bHi, __fmul_rn(wt, bmH1));

        const bool indLo = c1 > c0;                // argmax: first wins ties
        const bool indHi = c3 > c2;
        float nLo = fmaxf(c0, c1);
        float nHi = fmaxf(c2, c3);

        // mean over 64 states: butterfly reduction across the wave
        float s = nLo + nHi;
        s += __shfl_xor(s, 16, 32);
        s += __shfl_xor(s, 8, 32);
        s += __shfl_xor(s, 4, 32);
        s += __shfl_xor(s, 2, 32);
        s += __shfl_xor(s, 1, 32);
        const float mean = s * (1.0f / 64.0f);

        nLo = fminf(fmaxf(nLo - mean, -CLAMP_V), CLAMP_V);
        nHi = fminf(fmaxf(nHi - mean, -CLAMP_V), CLAMP_V);
        plo = nLo;
        phi = nHi;

        // decision bits -> wave32 ballots -> LDS (only t >= 1024 needed)
        const unsigned int mLo = __builtin_amdgcn_ballot_w32(indLo);
        const unsigned int mHi = __builtin_amdgcn_ballot_w32(indHi);
        if (t >= DET_LEN) {
            if (lane < 2) msk[((t - DET_LEN) << 1) + lane] = lane ? mHi : mLo;
        }

        // soft output window: out_probs[1023:2048]
        if (t >= DET_LEN - 1 && t < 2 * DET_LEN) {
            float* p = softb + (size_t)(t - (DET_LEN - 1)) * NSTATES;
            p[lane]      = nLo;
            p[lane + 32] = nHi;
        }
    }

    __syncthreads();

    // ---- Traceback (sequential in state; LDS-resident decisions) -----------
    if (lane == 0) {
        int state = 0;                              // state_{T} = 0
        float* db = dec_out + (size_t)b * DET_LEN;
        for (int t = T_TOT - 1; t >= DET_LEN; --t) {
            // bit_t = (state_{t+1} + 1) % 2, emitted before following prev
            if (t < 2 * DET_LEN) db[t - DET_LEN] = (float)((state + 1) & 1);
            const unsigned int m = msk[((t - DET_LEN) << 1) + (state >> 5)];
            const int ind = (m >> (state & 31)) & 1;
            state = (state >> 1) + (ind << 5);      // TRANS_PREV[state, ind]
        }
    }
}

extern "C" void kernel_launch(void* const* d_in, const int* in_sizes, int n_in,
                              void* d_out, int out_size, void* d_ws, size_t ws_size,
                              hipStream_t stream) {
    const float* x   = (const float*)d_in[0];   // (256, 2048) f32
    const float* wts = (const float*)d_in[1];   // (3072,)     f32
    float* dec  = (float*)d_out;                               // (256,1024)
    float* soft = (float*)d_out + (size_t)BATCH * DET_LEN;     // (256,1025*64)

    hipLaunchKernelGGL(wcva_decode_kernel, dim3(BATCH), dim3(32), 0, stream,
                       x, wts, dec, soft);
}